// FreTransferInv_25864293056756
// MI455X (gfx1250) — compile-verified
//
#include <hip/hip_runtime.h>

// Inverse Haar-style conv_transpose2d(stride=2, k=2x2, block-diagonal filter).
// x: (16,16,256,256) f32  ->  out: (16,4,512,512) f32
// out[b, j, 2h+a, 2w+c] = sum_i x[b, 4i+j, h, w] * F_i[a][c],
//   F order: [g00, g10, g01, g11]  (filters2 order in the reference).
//
// Memory-bound streaming kernel (AI ~0.5 FLOP/B): staged through LDS with
// CDNA5 async global->LDS copies (ASYNCcnt) to decouple loads from VGPRs.

namespace {

constexpr int BATCH = 16;
constexpr int CIN   = 16;
constexpr int HIN   = 256;
constexpr int WIN   = 256;
constexpr int NOUT  = 4;
constexpr int HOUT  = 512;
constexpr int WOUT  = 512;
constexpr int ROWS  = 4;          // input rows per workgroup

__global__ __launch_bounds__(256)
void haar_inv_kernel(const float* __restrict__ x,
                     const float* __restrict__ g00,
                     const float* __restrict__ g01,
                     const float* __restrict__ g10,
                     const float* __restrict__ g11,
                     float* __restrict__ out)
{
    // [subband i][row r][w]  = 4 * 4 * 256 floats = 16 KB
    __shared__ float smem[4 * ROWS * WIN];

    const int tid  = threadIdx.x;          // 0..255
    const int blk  = blockIdx.x;           // 0..4095
    const int hBlk = blk & 63;             // 64 row-blocks of 4
    const int j    = (blk >> 6) & 3;       // output channel
    const int b    = blk >> 8;             // batch
    const int h0   = hBlk * ROWS;

    const int r  = (tid >> 6) & 3;         // row within block (0..3)
    const int w4 = tid & 63;               // float4 index within row (0..63)

    // Low 32 bits of the generic pointer to __shared__ data == LDS byte offset
    // within the workgroup allocation (what GLOBAL_LOAD_ASYNC_TO_LDS wants).
    const unsigned ldsBase = (unsigned)(size_t)(&smem[0]);

    // ---- Stage 1: async global -> LDS (4 x B128 per thread = 16 KB/block) ----
#pragma unroll
    for (int k = 0; k < 4; ++k) {          // k = sub-band index
        const unsigned gOff = (unsigned)(
            ((((b * CIN + (4 * k + j)) * HIN) + (h0 + r)) * WIN + w4 * 4) *
            (int)sizeof(float));
        const unsigned ldsOff = ldsBase + (unsigned)((k * 256 + tid) * 16);
        asm volatile("global_load_async_to_lds_b128 %0, %1, %2"
                     :
                     : "v"(ldsOff), "v"(gOff), "s"(x)
                     : "memory");
    }
    asm volatile("s_wait_asynccnt 0" ::: "memory");
    __syncthreads();

    // ---- Filter taps (uniform; tiny, L2-resident). filters2 = [g00,g10,g01,g11]
    float f0[4], f1[4], f2[4], f3[4];      // [a*2 + c]
#pragma unroll
    for (int t = 0; t < 4; ++t) {
        f0[t] = g00[t];
        f1[t] = g10[t];
        f2[t] = g01[t];
        f3[t] = g11[t];
    }

    // ---- Stage 2: read 4 pixels per sub-band from LDS (ds_load_b128) ----
    alignas(16) float v0[4], v1[4], v2[4], v3[4];
    *reinterpret_cast<float4*>(v0) =
        *reinterpret_cast<const float4*>(&smem[(0 * ROWS + r) * WIN + w4 * 4]);
    *reinterpret_cast<float4*>(v1) =
        *reinterpret_cast<const float4*>(&smem[(1 * ROWS + r) * WIN + w4 * 4]);
    *reinterpret_cast<float4*>(v2) =
        *reinterpret_cast<const float4*>(&smem[(2 * ROWS + r) * WIN + w4 * 4]);
    *reinterpret_cast<float4*>(v3) =
        *reinterpret_cast<const float4*>(&smem[(3 * ROWS + r) * WIN + w4 * 4]);

    // ---- Butterfly: 4 input pixels -> 2x8 output floats ----
    alignas(16) float top[8], bot[8];
#pragma unroll
    for (int p = 0; p < 4; ++p) {
        top[2 * p]     = f0[0] * v0[p] + f1[0] * v1[p] + f2[0] * v2[p] + f3[0] * v3[p];
        top[2 * p + 1] = f0[1] * v0[p] + f1[1] * v1[p] + f2[1] * v2[p] + f3[1] * v3[p];
        bot[2 * p]     = f0[2] * v0[p] + f1[2] * v1[p] + f2[2] * v2[p] + f3[2] * v3[p];
        bot[2 * p + 1] = f0[3] * v0[p] + f1[3] * v1[p] + f2[3] * v2[p] + f3[3] * v3[p];
    }

    // ---- Stores: 2 rows x 8 floats, 32B-aligned, lane-contiguous B128 ----
    const int oh = 2 * (h0 + r);
    const int ow = w4 * 8;
    float* o0 = out + (((size_t)(b * NOUT + j) * HOUT + oh) * WOUT + ow);
    float* o1 = o0 + WOUT;
    *reinterpret_cast<float4*>(o0)     = *reinterpret_cast<float4*>(&top[0]);
    *reinterpret_cast<float4*>(o0 + 4) = *reinterpret_cast<float4*>(&top[4]);
    *reinterpret_cast<float4*>(o1)     = *reinterpret_cast<float4*>(&bot[0]);
    *reinterpret_cast<float4*>(o1 + 4) = *reinterpret_cast<float4*>(&bot[4]);
}

} // namespace

extern "C" void kernel_launch(void* const* d_in, const int* in_sizes, int n_in,
                              void* d_out, int out_size, void* d_ws, size_t ws_size,
                              hipStream_t stream) {
    // setup_inputs order: x, g00, g01, g10, g11, n
    const float* x   = (const float*)d_in[0];
    const float* g00 = (const float*)d_in[1];
    const float* g01 = (const float*)d_in[2];
    const float* g10 = (const float*)d_in[3];
    const float* g11 = (const float*)d_in[4];
    float* out = (float*)d_out;

    const int nBlocks = BATCH * NOUT * (HIN / ROWS);   // 16*4*64 = 4096
    haar_inv_kernel<<<nBlocks, 256, 0, stream>>>(x, g00, g01, g10, g11, out);
}